// HungarianMatcher_14096082666122
// MI455X (gfx1250) — compile-verified
//
#include <hip/hip_runtime.h>
#include <hip/hip_bf16.h>
#include <math.h>

// ---------------- problem constants ----------------
#define NLAYERS 6
#define NB      32
#define NQ      300
#define NT      64
#define OC      151     // obj / sub classes (+1)
#define RC      51      // rel classes (+1)
#define ROWS    (NB*NQ)         // 9600 pred rows
#define COLS    (NB*NT)         // 2048 target cols
#define KTOT    (OC+OC+RC)      // 353 (valid k = 0..352)
#define KPAD    356             // multiple of 4 for K=4 WMMA steps
#define KSTEPS  (KPAD/4)        // 89
#define LAST_ROW_OFF ((NLAYERS-1)*ROWS)  // row offset of layer [-1]

typedef __attribute__((ext_vector_type(2))) float v2f;
typedef __attribute__((ext_vector_type(8))) float v8f;
typedef __attribute__((ext_vector_type(4))) int   v4i;

#define AS1 __attribute__((address_space(1)))
#define AS3 __attribute__((address_space(3)))

#if defined(__HIP_DEVICE_COMPILE__) && __has_builtin(__builtin_amdgcn_wmma_f32_16x16x4_f32)
#define HAS_WMMA4 1
#else
#define HAS_WMMA4 0
#endif

#if defined(__HIP_DEVICE_COMPILE__) && \
    __has_builtin(__builtin_amdgcn_global_load_async_to_lds_b128) && \
    __has_builtin(__builtin_amdgcn_s_wait_asynccnt)
#define HAS_ASYNC 1
#else
#define HAS_ASYNC 0
#endif

// ---------------- wave32 reductions ----------------
__device__ __forceinline__ float wave_max32(float v) {
  #pragma unroll
  for (int o = 16; o > 0; o >>= 1) v = fmaxf(v, __shfl_xor(v, o, 32));
  return v;
}
__device__ __forceinline__ float wave_sum32(float v) {
  #pragma unroll
  for (int o = 16; o > 0; o >>= 1) v += __shfl_xor(v, o, 32);
  return v;
}

// =====================================================================
// Kernel 1: per-row stable softmax for obj(151), sub(151), rel(51)
// Writes U[row][0..355]  (pads 353..355 zeroed).  One wave per row.
// =====================================================================
__device__ __forceinline__ void softmax_row(const float* __restrict__ src, int n,
                                            float* __restrict__ dst, int lane) {
  float x[5];
  float m = -INFINITY;
  #pragma unroll
  for (int i = 0; i < 5; ++i) {
    int k = lane + 32 * i;
    x[i] = (k < n) ? src[k] : -INFINITY;
    m = fmaxf(m, x[i]);
  }
  m = wave_max32(m);
  float s = 0.f;
  #pragma unroll
  for (int i = 0; i < 5; ++i) {
    int k = lane + 32 * i;
    if (k < n) { x[i] = expf(x[i] - m); s += x[i]; }
  }
  s = wave_sum32(s);
  #pragma unroll
  for (int i = 0; i < 5; ++i) {
    int k = lane + 32 * i;
    if (k < n) dst[k] = x[i] / s;
  }
}

__global__ void softmax_kernel(const float* __restrict__ obj_logits,
                               const float* __restrict__ sub_logits,
                               const float* __restrict__ rel_logits,
                               float* __restrict__ U) {
  const int wave = threadIdx.x >> 5;
  const int lane = threadIdx.x & 31;
  const int row  = blockIdx.x * 8 + wave;
  float* u = U + (size_t)row * KPAD;
  softmax_row(obj_logits + (size_t)(LAST_ROW_OFF + row) * OC, OC, u,        lane);
  softmax_row(sub_logits + (size_t)(LAST_ROW_OFF + row) * OC, OC, u + OC,   lane);
  softmax_row(rel_logits + (size_t)(LAST_ROW_OFF + row) * RC, RC, u + 2*OC, lane);
  if (lane < (KPAD - KTOT)) u[KTOT + lane] = 0.f;   // zero pad lanes
}

// =====================================================================
// Kernel 2: class cost  out[row,col] = -(U[row] . onehot(col))
// One-hot positions: p0=lab_obj, p1=151+lab_sub, p2=302+lab_rel.
// Exact in f32 (products are x*1.0 or x*0.0; exactly one hit per k-range).
// Block = 256 thr (8 waves) -> 16 rows x 128 cols; each wave: 16x16 tile
// via 89 x V_WMMA_F32_16X16X4_F32.  Grid = (COLS/128, ROWS/16).
// B one-hot built from a per-32-k bitmask: 2 bit-extracts + cvt per step.
//
// Assumed VGPR striping (per ISA 7.12.2, f32 K=4):
//   A (16x4): v0 = {K=0 | K=2}, v1 = {K=1 | K=3}, M = lane%16
//   B (4x16): v0 = {K=0 | K=2}, v1 = {K=1 | K=3}, N = lane%16 (mirror of A)
//   C (16x16): vr lanes0-15 -> M=r, lanes16-31 -> M=r+8, N = lane%16
// =====================================================================
__global__ void class_cost_kernel(const float* __restrict__ U,
                                  const int* __restrict__ lab_obj,
                                  const int* __restrict__ lab_sub,
                                  const int* __restrict__ lab_rel,
                                  float* __restrict__ out) {
  __shared__ __align__(16) float u[16 * KPAD];   // 22784 B
  const int tid = threadIdx.x;
  const float* src = U + (size_t)blockIdx.y * 16 * KPAD;
#if defined(__HIP_DEVICE_COMPILE__)
  __builtin_prefetch(src, 0, 0);                 // global_prefetch_b8
#endif
#if HAS_ASYNC
  // async global->LDS fill of the 16-row U tile (1424 x b128)
  for (int i4 = tid; i4 < (16 * KPAD) / 4; i4 += 256) {
    __builtin_amdgcn_global_load_async_to_lds_b128(
        (AS1 v4i*)(src + 4 * i4), (AS3 v4i*)(u + 4 * i4), 0, 0);
  }
  __builtin_amdgcn_s_wait_asynccnt(0);
  __syncthreads();
#else
  for (int i = tid; i < 16 * KPAD; i += 256) u[i] = src[i];
  __syncthreads();
#endif

  const int wave = tid >> 5;
  const int lane = tid & 31;
  const int n    = lane & 15;
  const int hi   = lane >> 4;                    // 0: lanes 0-15, 1: 16-31
  const int col  = blockIdx.x * 128 + wave * 16 + n;
  const int p0   = lab_obj[col];
  const int p1   = OC     + lab_sub[col];
  const int p2   = 2 * OC + lab_rel[col];

#if HAS_WMMA4
  v8f acc = {};
  const float* arow = u + n * KPAD + 2 * hi;     // this lane's A (M=n) K-pair base
  #pragma unroll 1
  for (int w = 0; w < 11; ++w) {                 // words 0..10: k = 32w..32w+31
    unsigned m = 0u;
    m |= ((p0 >> 5) == w) ? (1u << (p0 & 31)) : 0u;
    m |= ((p1 >> 5) == w) ? (1u << (p1 & 31)) : 0u;
    m |= ((p2 >> 5) == w) ? (1u << (p2 & 31)) : 0u;
    #pragma unroll
    for (int t = 0; t < 8; ++t) {                // step s = 8w + t
      const int koff = (w << 5) + 4 * t;
      v2f a; a.x = arow[koff]; a.y = arow[koff + 1];
      const unsigned sh = 4u * (unsigned)t + 2u * (unsigned)hi;
      v2f b;
      b.x = (float)((m >> sh) & 1u);
      b.y = (float)((m >> (sh + 1u)) & 1u);
      acc = __builtin_amdgcn_wmma_f32_16x16x4_f32(false, a, false, b,
                                                  (short)0, acc, false, false);
    }
  }
  { // tail step s=88 (k=352..355; only k=352 can be hot: p2 <= 352)
    unsigned m = ((p2 >> 5) == 11) ? (1u << (p2 & 31)) : 0u;
    v2f a; a.x = arow[352]; a.y = arow[353];
    const unsigned sh = 2u * (unsigned)hi;
    v2f b;
    b.x = (float)((m >> sh) & 1u);
    b.y = (float)((m >> (sh + 1u)) & 1u);
    acc = __builtin_amdgcn_wmma_f32_16x16x4_f32(false, a, false, b,
                                                (short)0, acc, false, false);
  }
  const int rbase = blockIdx.y * 16 + hi * 8;
  #pragma unroll
  for (int r = 0; r < 8; ++r)
    out[(size_t)(rbase + r) * COLS + col] = -acc[r];
#else
  // Exact LDS-gather fallback (no 16x16x4 f32 builtin on this toolchain)
  for (int j = tid; j < 16 * 128; j += 256) {
    const int r = j >> 7, c = j & 127;
    const int col2 = blockIdx.x * 128 + c;
    const int q0 = lab_obj[col2], q1 = OC + lab_sub[col2], q2 = 2*OC + lab_rel[col2];
    const float* ur = u + r * KPAD;
    out[(size_t)(blockIdx.y * 16 + r) * COLS + col2] = -(ur[q0] + ur[q1] + ur[q2]);
  }
#endif
}

// =====================================================================
// Kernel 3 (VALU workhorse): L1 + L2 + two GIoU terms, accumulated
// into out[].  Block = 256 thr = 8 waves; wave <-> one pred row;
// block covers 8 rows x 1024 cols.  Grid = (ROWS/8, COLS/1024).
// Raw target boxes pulled in via async global->LDS, then derived table
// (cxcywh + xyxy + area) built in LDS, stride 21 floats (odd multiple
// -> 64 banks conflict-free for lane-strided b32 reads).
// =====================================================================
__device__ __forceinline__ float giou_term(float x1, float y1, float x2, float y2, float a1,
                                           float tx1, float ty1, float tx2, float ty2, float a2) {
  const float ltx = fmaxf(x1, tx1), lty = fmaxf(y1, ty1);
  const float rbx = fminf(x2, tx2), rby = fminf(y2, ty2);
  const float w = fmaxf(rbx - ltx, 0.f), h = fmaxf(rby - lty, 0.f);
  const float inter = w * h;
  const float uni = a1 + a2 - inter;
  const float iou = inter / uni;
  const float ex1 = fminf(x1, tx1), ey1 = fminf(y1, ty1);
  const float ex2 = fmaxf(x2, tx2), ey2 = fmaxf(y2, ty2);
  const float ew = fmaxf(ex2 - ex1, 0.f), eh = fmaxf(ey2 - ey1, 0.f);
  const float ea = ew * eh;
  return iou - (ea - uni) / ea;
}

#define TSTRIDE 21
__global__ void box_cost_kernel(const float* __restrict__ pred_obj_box,
                                const float* __restrict__ pred_sub_box,
                                const float* __restrict__ tgt_obj_box,
                                const float* __restrict__ tgt_sub_box,
                                float* __restrict__ out) {
  __shared__ float T[1024 * TSTRIDE];            // 86016 B derived table
#if HAS_ASYNC
  __shared__ __align__(16) float Traw[2 * 1024 * 4];  // 32768 B raw boxes
#endif
  const int tid = threadIdx.x;
  const int colBase = blockIdx.y * 1024;
#if defined(__HIP_DEVICE_COMPILE__)
  __builtin_prefetch(tgt_obj_box + (size_t)colBase * 4, 0, 0);
  __builtin_prefetch(tgt_sub_box + (size_t)colBase * 4, 0, 0);
#endif
#if HAS_ASYNC
  for (int i = tid; i < 1024; i += 256) {
    __builtin_amdgcn_global_load_async_to_lds_b128(
        (AS1 v4i*)(tgt_obj_box + 4 * (size_t)(colBase + i)),
        (AS3 v4i*)(Traw + 4 * i), 0, 0);
    __builtin_amdgcn_global_load_async_to_lds_b128(
        (AS1 v4i*)(tgt_sub_box + 4 * (size_t)(colBase + i)),
        (AS3 v4i*)(Traw + 4096 + 4 * i), 0, 0);
  }
  __builtin_amdgcn_s_wait_asynccnt(0);
  __syncthreads();
#endif
  for (int i = tid; i < 1024; i += 256) {
#if HAS_ASYNC
    const float4 ob = ((const float4*)Traw)[i];
    const float4 sb = ((const float4*)Traw)[1024 + i];
#else
    const float4 ob = ((const float4*)tgt_obj_box)[colBase + i];
    const float4 sb = ((const float4*)tgt_sub_box)[colBase + i];
#endif
    float* t = T + i * TSTRIDE;
    // obj: cxcywh, xyxy, area (area from xyxy to match reference rounding)
    t[0] = ob.x; t[1] = ob.y; t[2] = ob.z; t[3] = ob.w;
    {
      const float x1 = ob.x - 0.5f * ob.z, y1 = ob.y - 0.5f * ob.w;
      const float x2 = ob.x + 0.5f * ob.z, y2 = ob.y + 0.5f * ob.w;
      t[4] = x1; t[5] = y1; t[6] = x2; t[7] = y2;
      t[8] = (x2 - x1) * (y2 - y1);
    }
    // sub
    t[9] = sb.x; t[10] = sb.y; t[11] = sb.z; t[12] = sb.w;
    {
      const float x1 = sb.x - 0.5f * sb.z, y1 = sb.y - 0.5f * sb.w;
      const float x2 = sb.x + 0.5f * sb.z, y2 = sb.y + 0.5f * sb.w;
      t[13] = x1; t[14] = y1; t[15] = x2; t[16] = y2;
      t[17] = (x2 - x1) * (y2 - y1);
    }
  }
  __syncthreads();

  const int wave = tid >> 5;
  const int lane = tid & 31;
  const int row  = blockIdx.x * 8 + wave;
  const float4 po = ((const float4*)pred_obj_box)[LAST_ROW_OFF + row];
  const float4 ps = ((const float4*)pred_sub_box)[LAST_ROW_OFF + row];
  const float pox1 = po.x - 0.5f * po.z, poy1 = po.y - 0.5f * po.w;
  const float pox2 = po.x + 0.5f * po.z, poy2 = po.y + 0.5f * po.w;
  const float poA  = (pox2 - pox1) * (poy2 - poy1);
  const float psx1 = ps.x - 0.5f * ps.z, psy1 = ps.y - 0.5f * ps.w;
  const float psx2 = ps.x + 0.5f * ps.z, psy2 = ps.y + 0.5f * ps.w;
  const float psA  = (psx2 - psx1) * (psy2 - psy1);

  float* orow = out + (size_t)row * COLS + colBase;
  #pragma unroll 4
  for (int i = 0; i < 32; ++i) {
    const int c = i * 32 + lane;
    const float* t = T + c * TSTRIDE;
    // obj: L1 on cxcywh
    const float l1 = fabsf(po.x - t[0]) + fabsf(po.y - t[1]) +
                     fabsf(po.z - t[2]) + fabsf(po.w - t[3]);
    const float g_o = giou_term(pox1, poy1, pox2, poy2, poA,
                                t[4], t[5], t[6], t[7], t[8]);
    // sub: L2 on cxcywh
    const float dx = ps.x - t[9], dy = ps.y - t[10];
    const float dz = ps.z - t[11], dw = ps.w - t[12];
    const float l2 = sqrtf(fmaxf(dx*dx + dy*dy + dz*dz + dw*dw, 1e-30f));
    const float g_s = giou_term(psx1, psy1, psx2, psy2, psA,
                                t[13], t[14], t[15], t[16], t[17]);
    orow[c] += l1 - g_o + l2 - g_s;
  }
}

// =====================================================================
extern "C" void kernel_launch(void* const* d_in, const int* in_sizes, int n_in,
                              void* d_out, int out_size, void* d_ws, size_t ws_size,
                              hipStream_t stream) {
  (void)in_sizes; (void)n_in; (void)out_size; (void)ws_size;
  const float* obj_logits = (const float*)d_in[0];
  const float* obj_boxes  = (const float*)d_in[1];
  const int*   t_obj_lab  = (const int*)  d_in[2];
  const float* t_obj_box  = (const float*)d_in[3];
  const float* sub_logits = (const float*)d_in[4];
  const float* sub_boxes  = (const float*)d_in[5];
  const int*   t_sub_lab  = (const int*)  d_in[6];
  const float* t_sub_box  = (const float*)d_in[7];
  const float* rel_logits = (const float*)d_in[8];
  const int*   t_rel_lab  = (const int*)  d_in[9];
  float* out = (float*)d_out;
  float* U   = (float*)d_ws;                     // ROWS*KPAD floats = 13.7 MB

  softmax_kernel<<<ROWS / 8, 256, 0, stream>>>(obj_logits, sub_logits, rel_logits, U);
  class_cost_kernel<<<dim3(COLS / 128, ROWS / 16), 256, 0, stream>>>(
      U, t_obj_lab, t_sub_lab, t_rel_lab, out);
  box_cost_kernel<<<dim3(ROWS / 8, COLS / 1024), 256, 0, stream>>>(
      obj_boxes, sub_boxes, t_obj_box, t_sub_box, out);
}